// RadixLinearAttention_3298534883974
// MI455X (gfx1250) — compile-verified
//
#include <hip/hip_runtime.h>
#include <math.h>

typedef float v2f __attribute__((ext_vector_type(2)));
typedef float v4f __attribute__((ext_vector_type(4)));
typedef float v8f __attribute__((ext_vector_type(8)));

#define KSTR 132  // stride for [128]-wide LDS tiles: even, 16B-aligned rows, 132%64=4 -> conflict-free
#define TSTR 132  // state (transposed) stride
#define USTR 20   // U (transposed) stride: 20%64 pattern conflict-free, 16B-aligned rows
#define PSTR 18   // P stride (even -> b64 A-frags)
#define VSTR 17   // Vb / Am stride (scalar access only)

__device__ __forceinline__ v8f wmma4(v2f a, v2f b, v8f c) {
  // D = A(16x4 f32) * B(4x16 f32) + C(16x16 f32)
  return __builtin_amdgcn_wmma_f32_16x16x4_f32(false, a, false, b, (short)0, c,
                                               false, false);
}

__device__ __forceinline__ v2f ld2(const float* p) { return *(const v2f*)p; }

// A fragment from row-major [M][stride]: element (m,k) = L[m*stride+k]; pair contiguous
__device__ __forceinline__ v2f ldA(const float* L, int stride, int m0, int ks, int lane) {
  int m  = m0 + (lane & 15);
  int kk = ks + ((lane >> 4) << 1);
  return ld2(L + m * stride + kk);
}
// B fragment where storage is transposed: element (k,n) = M[n*stride + k]; pair contiguous
__device__ __forceinline__ v2f ldBT(const float* M, int stride, int ks, int lane) {
  int n  = lane & 15;
  int kk = ks + ((lane >> 4) << 1);
  return ld2(M + n * stride + kk);
}
// A fragment of M^T: element (m,k) = M[k*stride + m] (strided pair)
__device__ __forceinline__ v2f ldAT(const float* M, int stride, int m0, int ks, int lane) {
  int m  = m0 + (lane & 15);
  int kk = ks + ((lane >> 4) << 1);
  v2f r; r.x = M[kk * stride + m]; r.y = M[(kk + 1) * stride + m]; return r;
}

__global__ __launch_bounds__(128) void gdr_chunk_kernel(
    const float* __restrict__ gq, const float* __restrict__ gk,
    const float* __restrict__ gv, const float* __restrict__ ga,
    const float* __restrict__ gb, const float* __restrict__ gAlog,
    const float* __restrict__ gdt, const float* __restrict__ gS0,
    float* __restrict__ go, float* __restrict__ gSf) {
  constexpr int T = 2048, H = 16, Dk = 128, Dv = 128, CH = 16, NC = T / CH;
  constexpr float scale = 0.08838834764831845f;  // Dv^-0.5

  __shared__ float sST [16 * TSTR];  // state TRANSPOSED: S[d][v] at sST[v*TSTR+d]
  __shared__ float sKbb[CH * KSTR];  // beta_t * b_t * k_t
  __shared__ float sKdv[CH * KSTR];  // k_t / b_t
  __shared__ float sQb [CH * KSTR];  // b_t * scale * q_t
  __shared__ float sUT [16 * USTR];  // U TRANSPOSED: U[t][v] at sUT[v*USTR+t]
  __shared__ float sAm [CH * VSTR];  // strict-lower gram
  __shared__ float sP  [CH * PSTR];  // lower-incl-diag QK gram
  __shared__ float sVb [CH * VSTR];  // beta_t * v_t (tile)
  __shared__ float sG[CH], sBeta[CH], sBexp[CH], sInvb[CH];
  __shared__ float sBlast;

  const int blk  = blockIdx.x;
  const int bb   = blk >> 7;        // batch
  const int rem  = blk & 127;
  const int h    = rem >> 3;        // head
  const int v0   = (rem & 7) << 4;  // Dv column-tile start
  const int tid  = threadIdx.x;
  const int lane = tid & 31;
  const int wv   = tid >> 5;

  const float Aexp = expf(gAlog[h]);
  const float dtb  = gdt[h];

  // init state tile (transposed) from state0
  {
    int d = tid >> 4, vv = tid & 15;
    for (int i = 0; i < 16; ++i, d += 8)
      sST[vv * TSTR + d] = gS0[(((size_t)bb * H + h) * Dk + d) * Dv + v0 + vv];
  }

  for (int ch = 0; ch < NC; ++ch) {
    const int t0 = ch * CH;
    __syncthreads();

    // per-token scalars
    if (tid < CH) {
      int t = tid;
      size_t gi = ((size_t)bb * T + t0 + t) * H + h;
      float x  = ga[gi] + dtb;
      float sp = (x > 20.f) ? x : log1pf(expf(x));
      sG[t]    = -Aexp * sp;
      sBeta[t] = 1.f / (1.f + expf(-gb[gi]));
    }
    __syncthreads();
    if (tid == 0) {
      float cum = 0.f;
      for (int t = 0; t < CH; ++t) {
        cum += sG[t];
        sBexp[t] = expf(cum);
        sInvb[t] = expf(-cum);
      }
      sBlast = sBexp[CH - 1];
    }
    __syncthreads();

    // stage chunk K/Q into LDS (b128 in, b128 out) with gates folded in
#pragma unroll
    for (int rr = 0; rr < 4; ++rr) {
      int idx = tid + (rr << 7);   // 0..511
      int t   = idx >> 5;          // token row
      int c4  = (idx & 31) << 2;   // column start (float4)
      size_t base = (((size_t)bb * T + t0 + t) * H + h) * (size_t)Dk + c4;
      v4f kv = *(const v4f*)(gk + base);
      v4f qv = *(const v4f*)(gq + base);
      float s1 = sBeta[t] * sBexp[t], s2 = sInvb[t], s3 = sBexp[t] * scale;
      *(v4f*)(sKbb + t * KSTR + c4) = kv * s1;
      *(v4f*)(sKdv + t * KSTR + c4) = kv * s2;
      *(v4f*)(sQb  + t * KSTR + c4) = qv * s3;
    }
    for (int i = tid; i < CH * 16; i += 128) {
      int t = i >> 4, n = i & 15;
      size_t base = (((size_t)bb * T + t0 + t) * H + h) * (size_t)Dv;
      sVb[t * VSTR + n] = sBeta[t] * gv[base + v0 + n];
    }
    if (t0 + CH < T) {  // prefetch next chunk's K/Q rows
      int t = tid >> 3, seg = (tid & 7) << 4;
      size_t base = (((size_t)bb * T + t0 + CH + t) * H + h) * (size_t)Dk + seg;
      __builtin_prefetch(&gk[base], 0, 0);
      __builtin_prefetch(&gq[base], 0, 0);
    }
    __syncthreads();

    // stage A: four independent 16x16 accumulations over K=128, one per wave
    const int nn = lane & 15;
    const int mb = (lane >> 4) << 3;
    v8f accWq = {0.f, 0.f, 0.f, 0.f, 0.f, 0.f, 0.f, 0.f};
    if (wv == 0) {  // Wk = Kbb @ S0 -> u_init = Vb - Wk (into transposed U)
      v8f acc = {0.f, 0.f, 0.f, 0.f, 0.f, 0.f, 0.f, 0.f};
      for (int ks = 0; ks < Dk; ks += 4)
        acc = wmma4(ldA(sKbb, KSTR, 0, ks, lane), ldBT(sST, TSTR, ks, lane), acc);
      v4f lo, hi;
#pragma unroll
      for (int r = 0; r < 4; ++r) {
        lo[r] = sVb[(mb + r) * VSTR + nn] - acc[r];
        hi[r] = sVb[(mb + 4 + r) * VSTR + nn] - acc[r + 4];
      }
      *(v4f*)(sUT + nn * USTR + mb)     = lo;
      *(v4f*)(sUT + nn * USTR + mb + 4) = hi;
    } else if (wv == 1) {  // Wq = Qb @ S0 (kept in registers)
      for (int ks = 0; ks < Dk; ks += 4)
        accWq = wmma4(ldA(sQb, KSTR, 0, ks, lane), ldBT(sST, TSTR, ks, lane), accWq);
    } else if (wv == 2) {  // A = Kbb @ Kdv^T, strict lower
      v8f acc = {0.f, 0.f, 0.f, 0.f, 0.f, 0.f, 0.f, 0.f};
      for (int ks = 0; ks < Dk; ks += 4)
        acc = wmma4(ldA(sKbb, KSTR, 0, ks, lane), ldBT(sKdv, KSTR, ks, lane), acc);
#pragma unroll
      for (int r = 0; r < 8; ++r) {
        int m = r + mb;
        sAm[m * VSTR + nn] = (nn < m) ? acc[r] : 0.f;
      }
    } else {  // P = Qb @ Kdv^T, lower incl diag
      v8f acc = {0.f, 0.f, 0.f, 0.f, 0.f, 0.f, 0.f, 0.f};
      for (int ks = 0; ks < Dk; ks += 4)
        acc = wmma4(ldA(sQb, KSTR, 0, ks, lane), ldBT(sKdv, KSTR, ks, lane), acc);
#pragma unroll
      for (int r = 0; r < 8; ++r) {
        int m = r + mb;
        sP[m * PSTR + nn] = (nn <= m) ? acc[r] : 0.f;
      }
    }
    __syncthreads();

    // triangular solve: U[t] = u_init[t] - sum_{i<t} A[t,i] U[i] (column-parallel)
    if (tid < 16) {
      int j = tid;
      float u[CH];
      v4f* uv = (v4f*)u;
#pragma unroll
      for (int r = 0; r < 4; ++r) uv[r] = *(const v4f*)(sUT + j * USTR + (r << 2));
      for (int t = 1; t < CH; ++t) {
        float acc2 = u[t];
        for (int i = 0; i < t; ++i) acc2 -= sAm[t * VSTR + i] * u[i];
        u[t] = acc2;
      }
#pragma unroll
      for (int r = 0; r < 4; ++r) *(v4f*)(sUT + j * USTR + (r << 2)) = uv[r];
    }
    __syncthreads();

    // stage D (wave 1): O = Wq + tril(P) @ U, write to global
    if (wv == 1) {
      for (int ks = 0; ks < CH; ks += 4)
        accWq = wmma4(ldA(sP, PSTR, 0, ks, lane), ldBT(sUT, USTR, ks, lane), accWq);
#pragma unroll
      for (int r = 0; r < 8; ++r) {
        int m = r + mb;
        go[(((size_t)bb * T + t0 + m) * H + h) * (size_t)Dv + v0 + nn] = accWq[r];
      }
    }

    // stage E (all waves): S = b_last * (S0 + Kdv^T @ U); wave owns rows [wv*32, wv*32+32)
    float bl = sBlast;
#pragma unroll
    for (int sub = 0; sub < 2; ++sub) {
      int m0 = ((wv << 1) + sub) << 4;
      int cb = nn * TSTR + m0 + mb;  // 8 consecutive state elems per lane
      v4f lo = *(const v4f*)(sST + cb);
      v4f hi = *(const v4f*)(sST + cb + 4);
      v8f c;
#pragma unroll
      for (int r = 0; r < 4; ++r) { c[r] = lo[r]; c[r + 4] = hi[r]; }
      for (int ks = 0; ks < CH; ks += 4)
        c = wmma4(ldAT(sKdv, KSTR, m0, ks, lane), ldBT(sUT, USTR, ks, lane), c);
#pragma unroll
      for (int r = 0; r < 4; ++r) { lo[r] = bl * c[r]; hi[r] = bl * c[r + 4]; }
      *(v4f*)(sST + cb)     = lo;
      *(v4f*)(sST + cb + 4) = hi;
    }
  }

  __syncthreads();
  // final state out (de-transpose)
  {
    int d = tid >> 4, vv = tid & 15;
    for (int i = 0; i < 16; ++i, d += 8)
      gSf[(((size_t)bb * H + h) * Dk + d) * Dv + v0 + vv] = sST[vv * TSTR + d];
  }
}

extern "C" void kernel_launch(void* const* d_in, const int* in_sizes, int n_in,
                              void* d_out, int out_size, void* d_ws, size_t ws_size,
                              hipStream_t stream) {
  (void)in_sizes; (void)n_in; (void)out_size; (void)d_ws; (void)ws_size;
  const float* gq    = (const float*)d_in[0];
  const float* gk    = (const float*)d_in[1];
  const float* gv    = (const float*)d_in[2];
  const float* ga    = (const float*)d_in[3];
  const float* gb    = (const float*)d_in[4];
  const float* gAlog = (const float*)d_in[5];
  const float* gdt   = (const float*)d_in[6];
  const float* gS0   = (const float*)d_in[7];

  float* go  = (float*)d_out;
  float* gSf = go + (size_t)2 * 2048 * 16 * 128;  // o first, then S_final

  dim3 grid(256);   // B * H * (Dv/16) = 2*16*8
  dim3 block(128);  // 4 wave32s
  gdr_chunk_kernel<<<grid, block, 0, stream>>>(gq, gk, gv, ga, gb, gAlog, gdt,
                                               gS0, go, gSf);
}